// SOMGrid_17987323035723
// MI455X (gfx1250) — compile-verified
//
#include <hip/hip_runtime.h>

typedef float v2f __attribute__((ext_vector_type(2)));
typedef float v8f __attribute__((ext_vector_type(8)));

#define SOM_H 32
#define SOM_W 32
#define SOM_HW 1024
#define SOM_DIM 512
#define SOM_BATCH 64
#define SOM_ND 2

// ---------------------------------------------------------------------------
// Kernel A: channel-sum reduction  s[b,hw] = sum_d x[b,d,hw]
// Expressed as D = A(ones,16x4) * B(x-slab,4x16) + C with V_WMMA_F32_16X16X4_F32,
// accumulating K=4 channels per WMMA, 128 WMMAs per 16-column spatial tile.
// f32 A/B keeps bit-accuracy vs. the f32 reference (no f16 rounding).
//
// B-matrix (4x16 f32) VGPR layout (ISA 7.12.2, 32-bit operand):
//   VGPR0: lanes 0-15 -> (K=0, N=lane), lanes 16-31 -> (K=2, N=lane-16)
//   VGPR1: lanes 0-15 -> (K=1, N=lane), lanes 16-31 -> (K=3, N=lane-16)
// so lane loads x[b, d+rowbase+0, col] and x[b, d+rowbase+1, col],
// rowbase = 0 for lanes 0-15, 2 for lanes 16-31.
// ---------------------------------------------------------------------------
__global__ __launch_bounds__(256) void som_reduce_wmma(
    const float* __restrict__ x, float* __restrict__ s) {
  const int wave = (blockIdx.x * blockDim.x + threadIdx.x) >> 5;  // global wave id
  const int lane = threadIdx.x & 31;
  const int b    = wave >> 6;   // 64 tiles per batch
  const int tile = wave & 63;   // 16 spatial columns per tile

  const int rowbase = (lane >> 4) << 1;  // 0 or 2
  const float* p = x + (size_t)b * SOM_DIM * SOM_HW + tile * 16 + (lane & 15);

  v2f a;
  a.x = 1.0f;
  a.y = 1.0f;
  v8f acc = {};

#pragma unroll 8
  for (int d = 0; d < SOM_DIM; d += 4) {
    v2f bv;
    bv.x = p[(size_t)(d + rowbase) * SOM_HW];
    bv.y = p[(size_t)(d + rowbase + 1) * SOM_HW];
    // D = A*B + C : all 16 rows of D equal the 512-deep column sums
    acc = __builtin_amdgcn_wmma_f32_16x16x4_f32(
        /*neg_a=*/false, a, /*neg_b=*/false, bv,
        /*c_mod=*/(short)0, acc, /*reuse_a=*/false, /*reuse_b=*/false);
  }

  // acc VGPR0, lanes 0-15 hold (M=0, N=lane) = sum for column tile*16+lane
  if (lane < 16) {
    s[b * SOM_HW + tile * 16 + lane] = acc[0];
  }
}

// ---------------------------------------------------------------------------
// Kernel B: 5x5 zero-padded blur of s[b] (staged in LDS with halo), then
// broadcast the blurred 32x32 map across 32 dim-rows per block with
// coalesced float4 (global_store_b128) stores. grid = BATCH * (DIM/32).
// ---------------------------------------------------------------------------
__global__ __launch_bounds__(256) void som_blur_broadcast(
    const float* __restrict__ s, const float* __restrict__ kern,
    float* __restrict__ out) {
  __shared__ float  halo[SOM_H + 4][SOM_W + 4];   // 36x36, zero halo
  __shared__ float4 blur4[SOM_HW / 4];            // blurred 32x32
  float* blurred = (float*)blur4;

  const int tid    = threadIdx.x;
  const int b      = blockIdx.x >> 4;   // /16 dim-chunks
  const int dchunk = blockIdx.x & 15;   // 32 dims per chunk

  // zero the halo frame, then fill the interior
  for (int i = tid; i < (SOM_H + 4) * (SOM_W + 4); i += 256)
    ((float*)halo)[i] = 0.0f;
  __syncthreads();
  for (int i = tid; i < SOM_HW; i += 256)
    halo[(i >> 5) + SOM_ND][(i & 31) + SOM_ND] = s[b * SOM_HW + i];

  float k[25];
#pragma unroll
  for (int i = 0; i < 25; ++i) k[i] = kern[i];  // uniform -> scalar loads
  __syncthreads();

  for (int p = tid; p < SOM_HW; p += 256) {
    const int r = p >> 5, c = p & 31;
    float acc = 0.0f;
#pragma unroll
    for (int ky = 0; ky < 5; ++ky)
#pragma unroll
      for (int kx = 0; kx < 5; ++kx)
        acc += halo[r + ky][c + kx] * k[ky * 5 + kx];
    blurred[p] = acc;
  }
  __syncthreads();

  // broadcast: 32 identical rows of 1024 floats, written as float4
  float4* o = (float4*)(out + (size_t)(b * SOM_DIM + dchunk * 32) * SOM_HW);
  for (int i = tid; i < 32 * (SOM_HW / 4); i += 256)
    o[i] = blur4[i & (SOM_HW / 4 - 1)];
}

extern "C" void kernel_launch(void* const* d_in, const int* in_sizes, int n_in,
                              void* d_out, int out_size, void* d_ws, size_t ws_size,
                              hipStream_t stream) {
  (void)in_sizes; (void)n_in; (void)out_size; (void)ws_size;
  const float* x    = (const float*)d_in[0];  // [64, 512, 1024] f32
  const float* kern = (const float*)d_in[1];  // [5, 5] f32
  float* out = (float*)d_out;                 // [64, 512, 1024] f32
  float* s   = (float*)d_ws;                  // [64, 1024] f32 scratch (256 KB)

  // A: 64 batches * 64 tiles = 4096 waves, 8 waves (256 threads) per block
  som_reduce_wmma<<<(SOM_BATCH * 64) / 8, 256, 0, stream>>>(x, s);
  // B: 64 batches * 16 dim-chunks
  som_blur_broadcast<<<SOM_BATCH * 16, 256, 0, stream>>>(s, kern, out);
}